// SimpleProteinGNN_20392504721598
// MI455X (gfx1250) — compile-verified
//
#include <hip/hip_runtime.h>
#include <hip/hip_bf16.h>

// ---------------------------------------------------------------------------
// GAT on MI455X (gfx1250). fp32 WMMA (V_WMMA_F32_16X16X4_F32) for the node
// transforms; L2-resident edge gather + fp32 atomic scatter for aggregation.
// ---------------------------------------------------------------------------

typedef __attribute__((ext_vector_type(2))) float v2f;
typedef __attribute__((ext_vector_type(8))) float v8f;

#define NEG_SLOPE 0.2f

// ---------------------------------------------------------------------------
// GEMM: H[N,64] = X[N,K] @ W[K,64] using V_WMMA_F32_16X16X4_F32.
// One wave per 16-row tile; 4 accumulators cover the 64 output channels.
// A-frag (16x4 f32): lanes 0-15 hold K=0,1 ; lanes 16-31 hold K=2,3 (row = lane%16)
// B-frag (4x16 f32): lanes 0-15 hold K=0,1 ; lanes 16-31 hold K=2,3 (col = lane%16)
// C/D (16x16 f32):   vgpr r of lane L holds M = r + 8*(L/16), Ncol = L%16
// ---------------------------------------------------------------------------
__global__ __launch_bounds__(128) void gat_gemm_wmma(
    const float* __restrict__ X, const float* __restrict__ W,
    float* __restrict__ H, int nNodes, int K)
{
    const int lane = threadIdx.x & 31;
    const int wave = threadIdx.x >> 5;
    const int tile = blockIdx.x * 4 + wave;
    if (tile * 16 >= nNodes) return;           // wave-uniform exit

    const int half = lane >> 4;                // 0 or 1
    const int l16  = lane & 15;
    const int kh   = half * 2;

    int row = tile * 16 + l16;
    if (row >= nNodes) row = nNodes - 1;       // branchless clamp (stores guarded)

    v8f acc0 = {}, acc1 = {}, acc2 = {}, acc3 = {};

    for (int k = 0; k < K; k += 4) {
        v2f a;
        a.x = X[(size_t)row * K + k + kh];
        a.y = X[(size_t)row * K + k + kh + 1];

        const float* W0 = W + (size_t)(k + kh) * 64;
        const float* W1 = W0 + 64;
        v2f b0, b1, b2, b3;
        b0.x = W0[l16 +  0]; b0.y = W1[l16 +  0];
        b1.x = W0[l16 + 16]; b1.y = W1[l16 + 16];
        b2.x = W0[l16 + 32]; b2.y = W1[l16 + 32];
        b3.x = W0[l16 + 48]; b3.y = W1[l16 + 48];

        acc0 = __builtin_amdgcn_wmma_f32_16x16x4_f32(false, a, false, b0, (short)0, acc0, false, false);
        acc1 = __builtin_amdgcn_wmma_f32_16x16x4_f32(false, a, false, b1, (short)0, acc1, false, false);
        acc2 = __builtin_amdgcn_wmma_f32_16x16x4_f32(false, a, false, b2, (short)0, acc2, false, false);
        acc3 = __builtin_amdgcn_wmma_f32_16x16x4_f32(false, a, false, b3, (short)0, acc3, false, false);
    }

    const int m0 = half * 8;
#pragma unroll
    for (int r = 0; r < 8; ++r) {
        int mrow = tile * 16 + m0 + r;
        if (mrow < nNodes) {
            float* Hr = H + (size_t)mrow * 64 + l16;
            Hr[ 0] = acc0[r];
            Hr[16] = acc1[r];
            Hr[32] = acc2[r];
            Hr[48] = acc3[r];
        }
    }
}

// ---------------------------------------------------------------------------
// Per-node attention scores: alpha_s[n] = h[n,:].a_src ; alpha_d[n] = h[n,:].a_dst
// ---------------------------------------------------------------------------
__global__ __launch_bounds__(256) void gat_alpha(
    const float* __restrict__ H,
    const float* __restrict__ a_s, const float* __restrict__ a_d,
    float* __restrict__ as_out, float* __restrict__ ad_out, int nNodes)
{
    int n = blockIdx.x * blockDim.x + threadIdx.x;
    if (n >= nNodes) return;
    const float4* hp  = (const float4*)(H + (size_t)n * 64);
    const float4* asp = (const float4*)a_s;
    const float4* adp = (const float4*)a_d;
    float s = 0.f, d = 0.f;
#pragma unroll
    for (int i = 0; i < 16; ++i) {
        float4 h4 = hp[i], s4 = asp[i], d4 = adp[i];
        s += h4.x * s4.x + h4.y * s4.y + h4.z * s4.z + h4.w * s4.w;
        d += h4.x * d4.x + h4.y * d4.y + h4.z * d4.z + h4.w * d4.w;
    }
    as_out[n] = s;
    ad_out[n] = d;
}

// ---------------------------------------------------------------------------
// Init: zero aggregation buffer, m = -inf, denom = 0
// ---------------------------------------------------------------------------
__global__ __launch_bounds__(256) void gat_init(
    float* __restrict__ agg, float* __restrict__ mmax, float* __restrict__ denom, int nNodes)
{
    long long idx = (long long)blockIdx.x * blockDim.x + threadIdx.x;
    long long tot = (long long)nNodes * 64;
    if (idx < tot) agg[idx] = 0.f;
    if (idx < nNodes) {
        mmax[idx]  = __int_as_float(0xFF800000);   // -inf
        denom[idx] = 0.f;
    }
}

__device__ __forceinline__ void edge_sd(const long long* __restrict__ ei,
                                        int e, int E, int& s, int& d)
{
    if (e < E) { s = (int)ei[e]; d = (int)ei[(size_t)E + e]; }
    else       { s = d = e - E; }                  // implicit self-loop
}

__device__ __forceinline__ float leaky(float v) {
    return v > 0.f ? v : NEG_SLOPE * v;
}

// IEEE-safe float atomic max (signed compare for >=0, unsigned min for <0)
__device__ __forceinline__ void atomicMaxF(float* addr, float val) {
    if (val >= 0.f) atomicMax((int*)addr, __float_as_int(val));
    else            atomicMin((unsigned int*)addr, __float_as_uint(val));
}

// Pass 1: segment max of leaky_relu(alpha_s[src] + alpha_d[dst]) over dst
__global__ __launch_bounds__(256) void gat_edge_max(
    const long long* __restrict__ ei,
    const float* __restrict__ as, const float* __restrict__ ad,
    float* __restrict__ mmax, int E, int N)
{
    int e = blockIdx.x * blockDim.x + threadIdx.x;
    if (e >= E + N) return;
    int s, d; edge_sd(ei, e, E, s, d);
    atomicMaxF(mmax + d, leaky(as[s] + ad[d]));
}

// Pass 2: ex = exp(e - m[dst]); att[e] = ex; denom[dst] += ex
__global__ __launch_bounds__(256) void gat_edge_expsum(
    const long long* __restrict__ ei,
    const float* __restrict__ as, const float* __restrict__ ad,
    const float* __restrict__ mmax, float* __restrict__ denom,
    float* __restrict__ att, int E, int N)
{
    int e = blockIdx.x * blockDim.x + threadIdx.x;
    if (e >= E + N) return;
    int s, d; edge_sd(ei, e, E, s, d);
    float ex = __expf(leaky(as[s] + ad[d]) - mmax[d]);
    att[e] = ex;
    atomicAdd(denom + d, ex);
}

// Pass 3: agg[dst,:] += (att[e]/denom[dst]) * h[src,:]
// 256-thread block = 4 edges x 64 channels; edge weight computed once into LDS.
__global__ __launch_bounds__(256) void gat_edge_agg(
    const long long* __restrict__ ei,
    const float* __restrict__ att, const float* __restrict__ denom,
    const float* __restrict__ h, float* __restrict__ agg, int E, int N)
{
    __shared__ int   s_sh[4];
    __shared__ int   d_sh[4];
    __shared__ float w_sh[4];
    const int Ep = E + N;

    if (threadIdx.x < 4) {
        int e = blockIdx.x * 4 + threadIdx.x;
        if (e < Ep) {
            int s, d; edge_sd(ei, e, E, s, d);
            s_sh[threadIdx.x] = s;
            d_sh[threadIdx.x] = d;
            w_sh[threadIdx.x] = att[e] / denom[d];
        } else {
            s_sh[threadIdx.x] = 0; d_sh[threadIdx.x] = 0; w_sh[threadIdx.x] = 0.f;
        }
    }
    __syncthreads();

    const int le = threadIdx.x >> 6;
    const int c  = threadIdx.x & 63;
    const int e  = blockIdx.x * 4 + le;
    if (e < Ep) {
        float w = w_sh[le];
        atomicAdd(agg + (size_t)d_sh[le] * 64 + c, w * h[(size_t)s_sh[le] * 64 + c]);
    }
}

// out = relu(agg + bias)  (layer 1 activation)
__global__ __launch_bounds__(256) void gat_bias_relu(
    float* __restrict__ agg, const float* __restrict__ b, int nNodes)
{
    long long idx = (long long)blockIdx.x * blockDim.x + threadIdx.x;
    if (idx >= (long long)nNodes * 64) return;
    float v = agg[idx] + b[idx & 63];
    agg[idx] = v > 0.f ? v : 0.f;
}

// d_out[c] = b2[c]   (fresh every launch -> deterministic under graph replay)
__global__ __launch_bounds__(64) void gat_out_init(
    float* __restrict__ out, const float* __restrict__ b2)
{
    out[threadIdx.x] = b2[threadIdx.x];
}

// d_out[c] += sum_n agg[n,c] / N   (chunked partial sums, 1 atomic per chunk)
__global__ __launch_bounds__(64) void gat_mean(
    const float* __restrict__ agg, float* __restrict__ out, int nNodes)
{
    const int c  = threadIdx.x;
    const int n0 = blockIdx.x * 512;
    const int n1 = min(n0 + 512, nNodes);
    float s = 0.f;
    for (int n = n0; n < n1; ++n) s += agg[(size_t)n * 64 + c];
    atomicAdd(out + c, s / (float)nNodes);
}

// ---------------------------------------------------------------------------
static void run_layer(const float* X, int K, const float* W,
                      const float* a_s, const float* a_d,
                      float* h, float* agg,
                      float* alpha_s, float* alpha_d,
                      float* mmax, float* denom, float* att,
                      const long long* ei, int N, int E, hipStream_t stream)
{
    const int tiles = (N + 15) / 16;
    gat_gemm_wmma<<<(tiles + 3) / 4, 128, 0, stream>>>(X, W, h, N, K);
    gat_alpha<<<(N + 255) / 256, 256, 0, stream>>>(h, a_s, a_d, alpha_s, alpha_d, N);

    long long tot = (long long)N * 64;
    gat_init<<<(int)((tot + 255) / 256), 256, 0, stream>>>(agg, mmax, denom, N);

    const int Ep = E + N;
    gat_edge_max   <<<(Ep + 255) / 256, 256, 0, stream>>>(ei, alpha_s, alpha_d, mmax, E, N);
    gat_edge_expsum<<<(Ep + 255) / 256, 256, 0, stream>>>(ei, alpha_s, alpha_d, mmax, denom, att, E, N);
    gat_edge_agg   <<<(Ep + 3) / 4,     256, 0, stream>>>(ei, att, denom, h, agg, E, N);
}

extern "C" void kernel_launch(void* const* d_in, const int* in_sizes, int n_in,
                              void* d_out, int out_size, void* d_ws, size_t ws_size,
                              hipStream_t stream)
{
    (void)n_in; (void)out_size; (void)ws_size;

    const float*     x    = (const float*)d_in[0];
    const long long* ei   = (const long long*)d_in[1];
    const float*     W1   = (const float*)d_in[2];
    const float*     as1  = (const float*)d_in[3];
    const float*     ad1  = (const float*)d_in[4];
    const float*     b1   = (const float*)d_in[5];
    const float*     W2   = (const float*)d_in[6];
    const float*     as2  = (const float*)d_in[7];
    const float*     ad2  = (const float*)d_in[8];
    const float*     b2   = (const float*)d_in[9];

    const int K1 = in_sizes[2] / 64;        // 20
    const int K2 = in_sizes[6] / 64;        // 64
    const int N  = in_sizes[0] / K1;        // 100000
    const int E  = in_sizes[1] / 2;         // 1000000

    float* ws      = (float*)d_ws;
    float* h_buf   = ws;                          // N*64  (h1, later h2)
    float* agg_buf = h_buf   + (size_t)N * 64;    // N*64  (agg1/out1, later agg2)
    float* alpha_s = agg_buf + (size_t)N * 64;    // N
    float* alpha_d = alpha_s + N;                 // N
    float* mmax    = alpha_d + N;                 // N
    float* denom   = mmax    + N;                 // N
    float* att     = denom   + N;                 // E + N

    // ---- Layer 1: h1 = x@W1 ; softmax-aggregate into agg_buf ; relu+bias ----
    run_layer(x, K1, W1, as1, ad1, h_buf, agg_buf,
              alpha_s, alpha_d, mmax, denom, att, ei, N, E, stream);
    gat_bias_relu<<<(int)(((long long)N * 64 + 255) / 256), 256, 0, stream>>>(agg_buf, b1, N);

    // ---- Layer 2: h2 = out1@W2 (overwrites h_buf) ; aggregate into agg_buf --
    run_layer(agg_buf, K2, W2, as2, ad2, h_buf, agg_buf,
              alpha_s, alpha_d, mmax, denom, att, ei, N, E, stream);

    // ---- Global mean pool: out = mean_n(agg2) + b2 --------------------------
    float* out = (float*)d_out;
    gat_out_init<<<1, 64, 0, stream>>>(out, b2);
    gat_mean<<<(N + 511) / 512, 64, 0, stream>>>(agg_buf, out, N);
}